// SSIMLoss_24206435680689
// MI455X (gfx1250) — compile-verified
//
#include <hip/hip_runtime.h>

typedef __attribute__((ext_vector_type(2))) float v2f;
typedef __attribute__((ext_vector_type(8))) float v8f;

#define H 256
#define Wd 256
#define OUTV 246              // valid output size (256-10)
#define NPLANES 496           // 16*31
#define RBLK 16               // row tiles of 16 (covers 246)
#define CBLK 4                // col tiles of 64 (covers 246)
#define NBLOCKS (NPLANES * RBLK * CBLK)   // 31744, one wave each
#define TW 64                 // output cols per wave
#define HALO_C 74             // input cols needed (64 + 10)
#define PADC 80               // padded cols: 5 vertical N-tiles of 16
#define STR 81                // LDS row stride (81 mod 64 banks is odd -> conflict-free)
#define NT_V 5                // vertical N-tiles
#define NT_H 4                // horizontal output tiles
#define C1 1.0e-4f
#define C2 9.0e-4f

__device__ __constant__ float c_g[11] = {
    0.00102838f, 0.00759876f, 0.03600077f, 0.10936070f, 0.21300553f,
    0.26601172f,
    0.21300553f, 0.10936070f, 0.03600077f, 0.00759876f, 0.00102838f};

__device__ __forceinline__ float bandw(int d) {
    return ((unsigned)d <= 10u) ? c_g[d] : 0.0f;   // g[d] if 0<=d<=10 else 0
}

__global__ __launch_bounds__(32)
void ssim_strip_kernel(const float* __restrict__ X, const float* __restrict__ Y,
                       float* __restrict__ partials) {
    // 5 channels x 28 halo rows x 81 floats = 45,360 B (single wave owns it all)
    __shared__ float s[5][28 * STR];

    const int lane = threadIdx.x;
    const int m    = lane & 15;   // A-frag row index / B-frag col index
    const int half = lane >> 4;   // K-striping half

    // block -> (plane, row tile, col tile); RBLK*CBLK == 64
    const int bid   = blockIdx.x;
    const int plane = bid >> 6;
    const int rem   = bid & 63;
    const int tr    = rem >> 2;
    const int tc    = rem & 3;
    const int row0  = tr * 16;
    const int col0  = tc * TW;
    const long plane_base = (long)plane * (H * Wd);

    // ---- Phase A: stage 26x74 halo of x,y; build 5 moment channels in LDS ----
    for (int r = 0; r < 28; ++r) {
        const int gr = row0 + r;
        for (int c = lane; c < PADC; c += 32) {
            const int gc = col0 + c;
            float xv = 0.0f, yv = 0.0f;
            if (r < 26 && c < HALO_C && gr < H && gc < Wd) {
                const long off = plane_base + (long)gr * Wd + gc;
                xv = X[off];
                yv = Y[off];
            }
            const int p = r * STR + c;
            s[0][p] = xv;
            s[1][p] = yv;
            s[2][p] = xv * xv;
            s[3][p] = yv * yv;
            s[4][p] = xv * yv;
        }
    }

    // ---- Band fragments: shared by vertical-A and horizontal-B ----
    // A(16x4): lane holds M=lane%16, VGPR v holds K = 4*kc + v + 2*half
    // B(4x16): lane holds N=lane%16, same K striping  -> identical values g[K-m]
    v2f bandA[7];
    #pragma unroll
    for (int kc = 0; kc < 7; ++kc) {
        const int k0 = kc * 4 + 2 * half;
        bandA[kc].x = bandw(k0 - m);
        bandA[kc].y = bandw(k0 + 1 - m);
    }

    // ---- Phase B: vertical pass per channel, V(16x80) = Gv(16x28 band) * In(28x80),
    //      written back over rows 0..15 of the same channel buffer ----
    for (int ch = 0; ch < 5; ++ch) {
        float* buf = &s[ch][0];
        v8f vacc[NT_V];
        #pragma unroll
        for (int nt = 0; nt < NT_V; ++nt) vacc[nt] = {};
        #pragma unroll
        for (int kc = 0; kc < 7; ++kc) {
            const int k0 = kc * 4 + 2 * half;   // input row for this lane-half
            #pragma unroll
            for (int nt = 0; nt < NT_V; ++nt) {
                v2f b;
                b.x = buf[k0 * STR + nt * 16 + m];
                b.y = buf[(k0 + 1) * STR + nt * 16 + m];
                vacc[nt] = __builtin_amdgcn_wmma_f32_16x16x4_f32(
                    false, bandA[kc], false, b, (short)0, vacc[nt], false, false);
            }
        }
        // D layout: vacc[nt][r] = (M = r + 8*half, N = nt*16 + m)
        #pragma unroll
        for (int nt = 0; nt < NT_V; ++nt) {
            #pragma unroll
            for (int r = 0; r < 8; ++r) {
                buf[(r + 8 * half) * STR + nt * 16 + m] = vacc[nt][r];
            }
        }
    }

    // ---- Phase C: horizontal pass + SSIM, 4 output tiles of 16x16 ----
    float part = 0.0f;
    for (int t = 0; t < NT_H; ++t) {
        v8f hacc[5];
        #pragma unroll
        for (int ch = 0; ch < 5; ++ch) hacc[ch] = {};
        #pragma unroll
        for (int kc = 0; kc < 7; ++kc) {
            const int k0 = kc * 4 + 2 * half;
            #pragma unroll
            for (int ch = 0; ch < 5; ++ch) {
                v2f a;   // A: M = lane%16 (row of V), K = halo col 16*t + k0 (+1)
                a.x = s[ch][m * STR + 16 * t + k0];
                a.y = s[ch][m * STR + 16 * t + k0 + 1];
                hacc[ch] = __builtin_amdgcn_wmma_f32_16x16x4_f32(
                    false, a, false, bandA[kc], (short)0, hacc[ch], false, false);
            }
        }
        // SSIM map for this 16x16 tile, masked to valid region
        #pragma unroll
        for (int r = 0; r < 8; ++r) {
            const int orow = row0 + r + 8 * half;
            const int ocol = col0 + 16 * t + m;
            const float mx  = hacc[0][r];
            const float my  = hacc[1][r];
            const float mxx = hacc[2][r];
            const float myy = hacc[3][r];
            const float mxy = hacc[4][r];
            const float mx2 = mx * mx, my2 = my * my;
            const float cov = mxy - mx * my;
            const float vx  = mxx - mx2;
            const float vy  = myy - my2;
            const float num = (2.0f * mx * my + C1) * (2.0f * cov + C2);
            const float den = (mx2 + my2 + C1) * (vx + vy + C2);
            const float v   = num / den;
            part += (orow < OUTV && ocol < OUTV) ? v : 0.0f;
        }
    }

    // ---- Phase D: wave32 reduce, one partial per block ----
    #pragma unroll
    for (int off = 16; off > 0; off >>= 1)
        part += __shfl_xor(part, off, 32);
    if (lane == 0) partials[bid] = part;
}

__global__ __launch_bounds__(256)
void ssim_finalize_kernel(const float* __restrict__ partials, int n,
                          float* __restrict__ out) {
    __shared__ float buf[256];
    float sum = 0.0f;
    for (int i = threadIdx.x; i < n; i += 256) sum += partials[i];
    buf[threadIdx.x] = sum;
    __syncthreads();
    for (int off = 128; off > 0; off >>= 1) {
        if ((int)threadIdx.x < off) buf[threadIdx.x] += buf[threadIdx.x + off];
        __syncthreads();
    }
    if (threadIdx.x == 0) {
        const float inv_count = 1.0f / (float)(NPLANES * OUTV * OUTV); // 30,015,936
        out[0] = 1.0f - buf[0] * inv_count;
    }
}

extern "C" void kernel_launch(void* const* d_in, const int* in_sizes, int n_in,
                              void* d_out, int out_size, void* d_ws, size_t ws_size,
                              hipStream_t stream) {
    const float* X = (const float*)d_in[0];   // outputs
    const float* Y = (const float*)d_in[1];   // label
    float* partials = (float*)d_ws;           // NBLOCKS floats (~127 KB)
    float* out = (float*)d_out;

    ssim_strip_kernel<<<NBLOCKS, 32, 0, stream>>>(X, Y, partials);
    ssim_finalize_kernel<<<1, 256, 0, stream>>>(partials, NBLOCKS, out);
}